// _PointnetSAModuleBase_1949915153006
// MI455X (gfx1250) — compile-verified
//
#include <hip/hip_runtime.h>
#include <hip/hip_bf16.h>
#include <math.h>

typedef float v2f __attribute__((ext_vector_type(2)));
typedef float v8f __attribute__((ext_vector_type(8)));

#define NB     8
#define NPTS   16384
#define NPT    512
#define NSAMP  32
#define RAD2   0.04f     // 0.2^2
#define BN_EPS 1e-5f

// ---------------------------------------------------------------------------
// 1) Furthest point sampling: one block per batch, 16 points/thread in regs.
// ---------------------------------------------------------------------------
__global__ __launch_bounds__(1024) void fps_kernel(const float* __restrict__ xyz,
                                                   int* __restrict__ fpsidx,
                                                   float* __restrict__ newxyz) {
  const int b   = blockIdx.x;
  const int tid = threadIdx.x;
  const float* base = xyz + (size_t)b * NPTS * 3;

  float px[16], py[16], pz[16], dd[16];
#pragma unroll
  for (int j = 0; j < 16; ++j) {
    int p = tid + j * 1024;
    px[j] = base[p * 3 + 0];
    py[j] = base[p * 3 + 1];
    pz[j] = base[p * 3 + 2];
    dd[j] = 3.4e38f;
  }

  __shared__ float rv[32];
  __shared__ int   ri[32];
  __shared__ int   s_last;

  const int wave = tid >> 5;
  const int lane = tid & 31;
  int last = 0;

  for (int i = 0; i < NPT; ++i) {
    if (tid == 0) {
      fpsidx[b * NPT + i] = last;
      newxyz[(b * NPT + i) * 3 + 0] = base[last * 3 + 0];
      newxyz[(b * NPT + i) * 3 + 1] = base[last * 3 + 1];
      newxyz[(b * NPT + i) * 3 + 2] = base[last * 3 + 2];
    }
    const float cx = base[last * 3 + 0];
    const float cy = base[last * 3 + 1];
    const float cz = base[last * 3 + 2];

    float bv = -1.0f;
    int   bi = 0x7fffffff;
#pragma unroll
    for (int j = 0; j < 16; ++j) {
      float dx = px[j] - cx, dy = py[j] - cy, dz = pz[j] - cz;
      float d  = dx * dx + dy * dy + dz * dz;
      float nd = fminf(dd[j], d);
      dd[j] = nd;
      if (nd > bv) { bv = nd; bi = tid + j * 1024; }   // ascending idx -> first-occurrence ties
    }
    // wave32 argmax reduce (lower index wins ties, matching jnp.argmax)
#pragma unroll
    for (int off = 16; off > 0; off >>= 1) {
      float ov = __shfl_xor(bv, off, 32);
      int   oi = __shfl_xor(bi, off, 32);
      if (ov > bv || (ov == bv && oi < bi)) { bv = ov; bi = oi; }
    }
    if (lane == 0) { rv[wave] = bv; ri[wave] = bi; }
    __syncthreads();
    if (wave == 0) {
      bv = rv[lane]; bi = ri[lane];
#pragma unroll
      for (int off = 16; off > 0; off >>= 1) {
        float ov = __shfl_xor(bv, off, 32);
        int   oi = __shfl_xor(bi, off, 32);
        if (ov > bv || (ov == bv && oi < bi)) { bv = ov; bi = oi; }
      }
      if (lane == 0) s_last = bi;
    }
    __syncthreads();
    last = s_last;
  }
}

// ---------------------------------------------------------------------------
// 2) Ball query: one wave32 per query; first NSAMP in-radius indices ascending.
// ---------------------------------------------------------------------------
__global__ __launch_bounds__(256) void ballq_kernel(const float* __restrict__ xyz,
                                                    const float* __restrict__ newxyz,
                                                    int* __restrict__ grp) {
  const int lane = threadIdx.x & 31;
  const int q    = blockIdx.x * 8 + (threadIdx.x >> 5);   // 512 blocks * 8 waves = 4096
  const int b    = q >> 9;
  const int s    = q & (NPT - 1);

  const float* base = xyz + (size_t)b * NPTS * 3;
  const float cx = newxyz[(b * NPT + s) * 3 + 0];
  const float cy = newxyz[(b * NPT + s) * 3 + 1];
  const float cz = newxyz[(b * NPT + s) * 3 + 2];

  int* g = grp + (size_t)q * NSAMP;
  int cnt = 0, first = -1;

  for (int bp = 0; bp < NPTS && cnt < NSAMP; bp += 32) {
    const int p = bp + lane;
    float dx = base[p * 3 + 0] - cx;
    float dy = base[p * 3 + 1] - cy;
    float dz = base[p * 3 + 2] - cz;
    bool hit = (dx * dx + dy * dy + dz * dz) <= RAD2;
    unsigned mask = (unsigned)__ballot(hit);
    if (mask) {
      if (first < 0) first = bp + __builtin_ctz(mask);
      int pos = __popc(mask & ((1u << lane) - 1u));
      if (hit && (cnt + pos) < NSAMP) g[cnt + pos] = p;
      cnt += __popc(mask);
    }
  }
  if (first < 0) first = NPTS - 1;           // all-miss: reference yields N-1
  int filled = cnt < NSAMP ? cnt : NSAMP;
  for (int slot = filled + lane; slot < NSAMP; slot += 32) g[slot] = first;
}

// ---------------------------------------------------------------------------
// 3) Grouped MLP via V_WMMA_F32_16X16X4_F32, one wave per query.
// ---------------------------------------------------------------------------
// D(32 x Nout) = relu( bn( A(32 x Kdim) * W^T ) ), W is [Nout][Kdim] row-major.
// A in LDS (stride lda), D to LDS (stride ldd). Kpad is Kdim rounded up to 4
// with A's padded columns zeroed so clamped weight reads contribute nothing.
__device__ __forceinline__ void gemm_bn_relu(const float* __restrict__ A, int lda,
                                             const float* __restrict__ W, int Kdim, int Kpad,
                                             int Nout,
                                             const float* __restrict__ gg,
                                             const float* __restrict__ bb,
                                             const float* __restrict__ mm,
                                             const float* __restrict__ vv,
                                             float* __restrict__ D, int ldd, int lane) {
  const int r15  = lane & 15;
  const int half = lane >> 4;
  for (int m0 = 0; m0 < 32; m0 += 16) {
    for (int n0 = 0; n0 < Nout; n0 += 16) {
      v8f acc = {0.f, 0.f, 0.f, 0.f, 0.f, 0.f, 0.f, 0.f};
      const int n = n0 + r15;
      for (int k0 = 0; k0 < Kpad; k0 += 4) {
        const int kk = k0 + half * 2;
        v2f a, w;
        // A 16x4 fragment: lanes 0-15 -> K = kk,kk+1 ; lanes 16-31 -> K+2
        a.x = A[(m0 + r15) * lda + kk];
        a.y = A[(m0 + r15) * lda + kk + 1];
        // B 4x16 fragment (W transposed on the fly); clamp keeps reads in
        // bounds — the matching A entries are zero-padded so result is exact.
        int ka = kk     < Kdim ? kk     : Kdim - 1;
        int kb = kk + 1 < Kdim ? kk + 1 : Kdim - 1;
        w.x = W[n * Kdim + ka];
        w.y = W[n * Kdim + kb];
        acc = __builtin_amdgcn_wmma_f32_16x16x4_f32(false, a, false, w,
                                                    (short)0, acc, false, false);
      }
      // BN fold: h*s + (b - m*s); whole D column of this lane is channel n.
      const float sc = gg[n] * rsqrtf(vv[n] + BN_EPS);
      const float bs = bb[n] - mm[n] * sc;
#pragma unroll
      for (int r = 0; r < 8; ++r) {
        float val = fmaxf(acc[r] * sc + bs, 0.f);
        D[(m0 + r + half * 8) * ldd + n] = val;
      }
    }
  }
}

__global__ __launch_bounds__(64) void mlp_kernel(const float* __restrict__ xyz,
                                                 const float* __restrict__ feat,
                                                 const int*   __restrict__ grp,
                                                 const float* __restrict__ newxyz,
                                                 const float* __restrict__ w0, const float* __restrict__ g0,
                                                 const float* __restrict__ b0, const float* __restrict__ m0,
                                                 const float* __restrict__ v0,
                                                 const float* __restrict__ w1, const float* __restrict__ g1,
                                                 const float* __restrict__ b1, const float* __restrict__ m1,
                                                 const float* __restrict__ v1,
                                                 const float* __restrict__ w2, const float* __restrict__ g2,
                                                 const float* __restrict__ b2, const float* __restrict__ m2,
                                                 const float* __restrict__ v2,
                                                 float* __restrict__ outfeat) {
  __shared__ float Xb[2][32 * 68];    // padded strides: 68, 132 -> conflict-free
  __shared__ float Hb[2][32 * 132];

  const int wave = threadIdx.x >> 5;
  const int lane = threadIdx.x & 31;
  const int q    = blockIdx.x * 2 + wave;
  const int b    = q >> 9;
  const int s    = q & (NPT - 1);

  float* X = Xb[wave];
  float* H = Hb[wave];

  // Build X: row = sample (lane), cols 0-2 = centered xyz, 3-66 = features, 67 = 0
  const int gi = grp[(size_t)q * NSAMP + lane];
  const float* prow = xyz + ((size_t)b * NPTS + gi) * 3;
  const float* nx   = newxyz + (b * NPT + s) * 3;
  float* xr = X + lane * 68;
  xr[0] = prow[0] - nx[0];
  xr[1] = prow[1] - nx[1];
  xr[2] = prow[2] - nx[2];
  const float4* frow = (const float4*)(feat + ((size_t)b * NPTS + gi) * 64);
#pragma unroll
  for (int c4 = 0; c4 < 16; ++c4) {
    float4 f = frow[c4];
    xr[3 + 4 * c4 + 0] = f.x;
    xr[3 + 4 * c4 + 1] = f.y;
    xr[3 + 4 * c4 + 2] = f.z;
    xr[3 + 4 * c4 + 3] = f.w;
  }
  xr[67] = 0.f;

  // layer0: X(32x67->68) -> H(32x64)
  gemm_bn_relu(X, 68,  w0, 67, 68, 64,  g0, b0, m0, v0, H, 132, lane);
  // layer1: H(32x64) -> X(32x64)
  gemm_bn_relu(H, 132, w1, 64, 64, 64,  g1, b1, m1, v1, X, 68,  lane);
  // layer2: X(32x64) -> H(32x128)
  gemm_bn_relu(X, 68,  w2, 64, 64, 128, g2, b2, m2, v2, H, 132, lane);

  // max over the 32 samples; output layout (B, 128, NPT)
#pragma unroll
  for (int cc = 0; cc < 4; ++cc) {
    const int col = lane + cc * 32;
    float mx = H[col];
    for (int row = 1; row < 32; ++row) mx = fmaxf(mx, H[row * 132 + col]);
    outfeat[((size_t)b * 128 + col) * NPT + s] = mx;
  }
}

// ---------------------------------------------------------------------------
extern "C" void kernel_launch(void* const* d_in, const int* in_sizes, int n_in,
                              void* d_out, int out_size, void* d_ws, size_t ws_size,
                              hipStream_t stream) {
  (void)in_sizes; (void)n_in; (void)out_size; (void)ws_size;
  const float* xyz  = (const float*)d_in[0];
  const float* feat = (const float*)d_in[1];
  const float* w0 = (const float*)d_in[2];
  const float* g0 = (const float*)d_in[3];
  const float* b0 = (const float*)d_in[4];
  const float* m0 = (const float*)d_in[5];
  const float* v0 = (const float*)d_in[6];
  const float* w1 = (const float*)d_in[7];
  const float* g1 = (const float*)d_in[8];
  const float* b1 = (const float*)d_in[9];
  const float* m1 = (const float*)d_in[10];
  const float* v1 = (const float*)d_in[11];
  const float* w2 = (const float*)d_in[12];
  const float* g2 = (const float*)d_in[13];
  const float* b2 = (const float*)d_in[14];
  const float* m2 = (const float*)d_in[15];
  const float* v2 = (const float*)d_in[16];

  float* newxyz  = (float*)d_out;                     // (8,512,3)
  float* outfeat = (float*)d_out + NB * NPT * 3;      // (8,128,512)

  int* fpsidx = (int*)d_ws;                           // 8*512 ints
  int* grp    = (int*)d_ws + NB * NPT;                // 8*512*32 ints

  fps_kernel  <<<NB,                1024, 0, stream>>>(xyz, fpsidx, newxyz);
  ballq_kernel<<<(NB * NPT) / 8,     256, 0, stream>>>(xyz, newxyz, grp);
  mlp_kernel  <<<(NB * NPT) / 2,      64, 0, stream>>>(xyz, feat, grp, newxyz,
                                                       w0, g0, b0, m0, v0,
                                                       w1, g1, b1, m1, v1,
                                                       w2, g2, b2, m2, v2,
                                                       outfeat);
}